// ResBlockEncoder_76467597738005
// MI455X (gfx1250) — compile-verified
//
#include <hip/hip_runtime.h>

// ResBlockEncoder: depthwise conv1d (C_IN=512 groups, 2 out-ch per group, K=3,
// stride 2, pad 1) + bias + stride-2 channel-duplicated identity residual.
//
// x:    [32, 512, 4095] f32      (d_in[0])
// W:    [1024, 1, 3]    f32      (d_in[1])
// bias: [1024]          f32      (d_in[2])
// out:  [32, 1024, 2048] f32     (d_out)
//
// out[b,oc,t] = w0*x[2t-1] + w1*x[2t] + w2*x[2t+1] + bias[oc] + x[2t],  ic=oc>>1
//
// Memory-bound (~537 MB total traffic, ~23 us at 23.3 TB/s). Strategy:
// async-DMA each input row into LDS via GVS-form GLOBAL_LOAD_ASYNC_TO_LDS_B32
// (scalar base + tid voff + imm offset -> no per-load 64-bit VALU address
// math; NT hint since data is single-use), aligned ds_load_b128 consumption,
// nontemporal b128 global stores.

#define B_DIM   32
#define C_IN    512
#define L_IN    4095
#define L_OUT   2048
#define THREADS 256

typedef __attribute__((ext_vector_type(4))) float v4f;

__global__ __launch_bounds__(THREADS) void resblock_dwconv_kernel(
    const float* __restrict__ x,
    const float* __restrict__ W,
    const float* __restrict__ bias,
    float* __restrict__ out)
{
    // Row staging buffer: 4095 data floats + 1 zero pad = 16 KB.
    __shared__ __align__(16) float smem[L_IN + 1];

    const int row = blockIdx.x;            // 0 .. B*C_IN-1
    const int b   = row >> 9;              // row / 512
    const int ic  = row & (C_IN - 1);      // row % 512
    const int tid = threadIdx.x;

    const float* __restrict__ xrow = x + (size_t)row * L_IN;

    // Right-edge zero pad (t = 2047 reads smem[4095]).
    if (tid == 0) smem[L_IN] = 0.0f;

    // -------- Async global -> LDS staging (CDNA5 path, ASYNCcnt) --------
    // GVS mode: mem = SADDR + VADDR + offset, lds = VDST + offset.
    // Rows are only 4B-aligned (4095*4 % 16 != 0), so b32 granularity is the
    // alignment-proof choice; each wave instruction still moves a contiguous
    // 128B span. NT hint: row is consumed once (no cross-block reuse).
    const unsigned lds0 = (unsigned)(size_t)(&smem[tid]);
    const unsigned voff = (unsigned)(tid * 4);

#define ALD(OFF) asm volatile(                                               \
        "global_load_async_to_lds_b32 %0, %1, %2 offset:" #OFF               \
        " th:TH_LOAD_NT"                                                     \
        :: "v"(lds0), "v"(voff), "s"(xrow) : "memory")

    ALD(0);     ALD(1024);  ALD(2048);  ALD(3072);
    ALD(4096);  ALD(5120);  ALD(6144);  ALD(7168);
    ALD(8192);  ALD(9216);  ALD(10240); ALD(11264);
    ALD(12288); ALD(13312); ALD(14336);
    if (tid < THREADS - 1) {               // idx 3840+tid < 4095
        ALD(15360);
    }
#undef ALD

    asm volatile("s_wait_asynccnt 0" ::: "memory");
    __syncthreads();

    // -------- Per-block uniform weights (scalar loads) --------
    const int oc0 = ic << 1;
    const float w00 = W[oc0 * 3 + 0];
    const float w01 = W[oc0 * 3 + 1];
    const float w02 = W[oc0 * 3 + 2];
    const float w10 = W[oc0 * 3 + 3];
    const float w11 = W[oc0 * 3 + 4];
    const float w12 = W[oc0 * 3 + 5];
    const float b0  = bias[oc0];
    const float b1  = bias[oc0 + 1];

    float* __restrict__ out0 = out + ((size_t)(b * 2 * C_IN + oc0)) * L_OUT;
    float* __restrict__ out1 = out0 + L_OUT;

    // -------- Compute: each thread -> 8 consecutive t for both oc --------
#pragma unroll
    for (int g = 0; g < 2; ++g) {
        const int t0 = tid * 8 + g * 4;        // multiple of 4 -> 32B LDS align

        // smem[2*t0 .. 2*t0+7] via two aligned b128 LDS loads + left tap.
        const v4f a = *(const v4f*)(&smem[2 * t0]);
        const v4f c = *(const v4f*)(&smem[2 * t0 + 4]);
        float pre = 0.0f;
        if (t0 > 0) pre = smem[2 * t0 - 1];    // left zero-pad only at t==0

        // taps for t0..t0+3:  xm = x[2t-1], x0 = x[2t], xp = x[2t+1]
        const float m0 = pre, c0 = a.x, p0 = a.y;
        const float m1 = a.y, c1 = a.z, p1 = a.w;
        const float m2 = a.w, c2 = c.x, p2 = c.y;
        const float m3 = c.y, c3 = c.z, p3 = c.w;

        v4f r0, r1;
        r0.x = fmaf(w00, m0, fmaf(w01, c0, fmaf(w02, p0, b0))) + c0;
        r0.y = fmaf(w00, m1, fmaf(w01, c1, fmaf(w02, p1, b0))) + c1;
        r0.z = fmaf(w00, m2, fmaf(w01, c2, fmaf(w02, p2, b0))) + c2;
        r0.w = fmaf(w00, m3, fmaf(w01, c3, fmaf(w02, p3, b0))) + c3;

        r1.x = fmaf(w10, m0, fmaf(w11, c0, fmaf(w12, p0, b1))) + c0;
        r1.y = fmaf(w10, m1, fmaf(w11, c1, fmaf(w12, p1, b1))) + c1;
        r1.z = fmaf(w10, m2, fmaf(w11, c2, fmaf(w12, p2, b1))) + c2;
        r1.w = fmaf(w10, m3, fmaf(w11, c3, fmaf(w12, p3, b1))) + c3;

        // Streaming output: written once, never re-read -> nontemporal b128.
        __builtin_nontemporal_store(r0, (v4f*)(out0 + t0));
        __builtin_nontemporal_store(r1, (v4f*)(out1 + t0));
    }
}

extern "C" void kernel_launch(void* const* d_in, const int* in_sizes, int n_in,
                              void* d_out, int out_size, void* d_ws, size_t ws_size,
                              hipStream_t stream) {
    const float* x    = (const float*)d_in[0];
    const float* W    = (const float*)d_in[1];
    const float* bias = (const float*)d_in[2];
    float* out        = (float*)d_out;

    const int nblocks = B_DIM * C_IN;          // 16384 rows, one WG each
    resblock_dwconv_kernel<<<nblocks, THREADS, 0, stream>>>(x, W, bias, out);
}